// EdgeFeatsLinear_3169685865351
// MI455X (gfx1250) — compile-verified
//
#include <hip/hip_runtime.h>

// Per-edge-type Linear(64->64) + ReLU, E=1M edges, 8 types.
// Strategy: bin edges by type (3 cheap passes into d_ws), then a grouped GEMM
// with persistent wave32 workers over the type-sorted tile list using
// V_WMMA_F32_16X16X4_F32. B operand (64x64 W of the current type) lives in
// 128 VGPRs and is reloaded only when a wave crosses a type boundary.
// Full 16-edge tiles (the ~100% case) take a scalar-branched fast path with
// unpredicated gathers/stores; only bin-tail tiles use per-lane masking.
//
// ws layout (int32 units):
//   [0..7]   per-type counts
//   [8..15]  per-type scatter cursors
//   [16..24] per-type exclusive offsets (9 entries)
//   [32..32+E) binned edge ids (order array)

typedef __attribute__((ext_vector_type(2))) float v2f;
typedef __attribute__((ext_vector_type(8))) float v8f;

#define NT 8
#define D 64

__global__ void EdgeBin_zero_kernel(int* __restrict__ ws) {
    int i = threadIdx.x;
    if (i < 16) ws[i] = 0;
}

__global__ void EdgeBin_count_kernel(const int* __restrict__ types,
                                     int* __restrict__ ws, int E) {
    int i = blockIdx.x * blockDim.x + threadIdx.x;
    if (i < E) {
        int t = types[i] & (NT - 1);
        atomicAdd(&ws[t], 1);
    }
}

__global__ void EdgeBin_scan_kernel(int* __restrict__ ws) {
    if (threadIdx.x == 0) {
        int acc = 0;
        for (int t = 0; t < NT; ++t) { ws[16 + t] = acc; acc += ws[t]; }
        ws[16 + NT] = acc;
    }
}

__global__ void EdgeBin_scatter_kernel(const int* __restrict__ types,
                                       int* __restrict__ ws, int E) {
    int i = blockIdx.x * blockDim.x + threadIdx.x;
    if (i < E) {
        int t = types[i] & (NT - 1);
        int p = atomicAdd(&ws[8 + t], 1);
        ws[32 + ws[16 + t] + p] = i;
    }
}

// Persistent wave32 workers: each wave owns a contiguous chunk of the
// type-sorted tile list; one tile = 16 edges of a single type.
__global__ void __launch_bounds__(128, 1)
EdgeFeatsLinear_wmma_kernel(const float* __restrict__ X,
                            const float* __restrict__ W,
                            const float* __restrict__ Bv,
                            const int* __restrict__ ws,
                            float* __restrict__ Y) {
    const int lane   = threadIdx.x & 31;
    const int wave   = (blockIdx.x * blockDim.x + threadIdx.x) >> 5;
    const int nwaves = (gridDim.x * blockDim.x) >> 5;

    const int* counts  = ws;
    const int* offsets = ws + 16;
    const int* order   = ws + 32;

    // Hoist bin metadata into registers (wave-uniform).
    int cnts[NT], offs[NT], tstart[NT];
    int total = 0;
#pragma unroll
    for (int t = 0; t < NT; ++t) {
        cnts[t]   = counts[t];
        offs[t]   = offsets[t];
        tstart[t] = total;
        total += (cnts[t] + 15) >> 4;
    }

    const int chunk = (total + nwaves - 1) / nwaves;
    const int tile0 = wave * chunk;
    const int tile1 = (tile0 + chunk < total) ? (tile0 + chunk) : total;
    if (tile0 >= tile1) return;  // wave-uniform exit

    // Lane roles (fixed across tiles).
    const int rowA  = lane & 15;        // A-fragment row
    const int kofs  = (lane >> 4) << 1; // K pair {0,1} lanes 0-15, {2,3} lanes 16-31
    const int mbase = (lane >> 4) << 3; // C/D rows 0..7 vs 8..15
    const int ncol  = lane & 15;        // B/C/D column within 16-block

    // Register-resident B operand + bias for the current type.
    v2f   bfr[4][16];
    float biasv[4];
    int   cur_t = -1;

    for (int tile = tile0; tile < tile1; ++tile) {
        // Map tile -> (type, local tile). tstart[] is monotonic.
        int t = 0;
#pragma unroll
        for (int ty = 1; ty < NT; ++ty)
            if (tile >= tstart[ty]) t = ty;
        const int local = tile - tstart[t];

        if (t != cur_t) {  // wave-uniform: reload W[t], b[t] into registers
            cur_t = t;
            const float* wt = W + (size_t)t * D * D;
#pragma unroll
            for (int nb = 0; nb < 4; ++nb) {
                const int col = nb * 16 + ncol;
                biasv[nb] = Bv[t * D + col];
#pragma unroll
                for (int kb = 0; kb < 16; ++kb) {
                    const int k0 = kb * 4 + kofs;
                    v2f bf;
                    bf.x = wt[(size_t)k0 * D + col];
                    bf.y = wt[(size_t)(k0 + 1) * D + col];
                    bfr[nb][kb] = bf;
                }
            }
        }

        // Force wave-uniform tile metadata into SGPRs -> scalar branches.
        const int  cnt  = __builtin_amdgcn_readfirstlane(cnts[t]);
        const int  base = __builtin_amdgcn_readfirstlane(offs[t] + local * 16);
        const bool fullTile =
            __builtin_amdgcn_readfirstlane((local * 16 + 16) <= cnt ? 1 : 0) != 0;

        // ---- Edge-id gathers ----
        int  eA;
        int  oidx[8];
        bool oval[8];
        if (fullTile) {
            eA = order[base + rowA];
#pragma unroll
            for (int r = 0; r < 8; ++r) {
                oidx[r] = order[base + mbase + r];
                oval[r] = true;
            }
        } else {
            const bool vA = (local * 16 + rowA) < cnt;
            eA = vA ? order[base + rowA] : 0;
#pragma unroll
            for (int r = 0; r < 8; ++r) {
                int m = mbase + r;
                bool v = (local * 16 + m) < cnt;
                oval[r] = v;
                oidx[r] = v ? order[base + m] : 0;
            }
        }

        // ---- A fragment gather: 16 rows x 64 cols fp32, streamed once (NT) ----
        const float* xrow = X + (size_t)eA * D;
        v2f afrag[16];
#pragma unroll
        for (int kb = 0; kb < 16; ++kb)
            afrag[kb] = __builtin_nontemporal_load(
                reinterpret_cast<const v2f*>(xrow + kb * 4 + kofs));

#pragma unroll
        for (int nb = 0; nb < 4; ++nb) {
            const int col = nb * 16 + ncol;
            const float bias = biasv[nb];
            v8f c = {bias, bias, bias, bias, bias, bias, bias, bias};
#pragma unroll
            for (int kb = 0; kb < 16; ++kb) {
                // 8 args: (neg_a, A, neg_b, B, c_mod, C, reuse_a, reuse_b)
                c = __builtin_amdgcn_wmma_f32_16x16x4_f32(
                        false, afrag[kb], false, bfr[nb][kb], (short)0, c,
                        false, false);
            }
            if (fullTile) {
#pragma unroll
                for (int r = 0; r < 8; ++r) {
                    float y = c[r];
                    y = y > 0.0f ? y : 0.0f;
                    __builtin_nontemporal_store(
                        y, &Y[(size_t)oidx[r] * D + col]);
                }
            } else {
#pragma unroll
                for (int r = 0; r < 8; ++r) {
                    if (oval[r]) {
                        float y = c[r];
                        y = y > 0.0f ? y : 0.0f;
                        __builtin_nontemporal_store(
                            y, &Y[(size_t)oidx[r] * D + col]);
                    }
                }
            }
        }
    }
}

extern "C" void kernel_launch(void* const* d_in, const int* in_sizes, int n_in,
                              void* d_out, int out_size, void* d_ws, size_t ws_size,
                              hipStream_t stream) {
    const float* X     = (const float*)d_in[0];   // edge_features [E,64] f32
    const int*   types = (const int*)d_in[1];     // edge_types [E]
    const float* W     = (const float*)d_in[2];   // [8,64,64] f32
    const float* Bv    = (const float*)d_in[3];   // [8,64] f32
    float* Y = (float*)d_out;                     // [E,64] f32
    int* ws  = (int*)d_ws;

    const int E = in_sizes[1];

    EdgeBin_zero_kernel<<<1, 32, 0, stream>>>(ws);

    const int blocks = (E + 255) / 256;
    EdgeBin_count_kernel<<<blocks, 256, 0, stream>>>(types, ws, E);
    EdgeBin_scan_kernel<<<1, 32, 0, stream>>>(ws);
    EdgeBin_scatter_kernel<<<blocks, 256, 0, stream>>>(types, ws, E);

    // Persistent GEMM workers: 2048 blocks x 4 waves = 8192 wave32 workers,
    // each owning a contiguous ~8-tile chunk of the type-sorted tile list.
    EdgeFeatsLinear_wmma_kernel<<<2048, 128, 0, stream>>>(X, W, Bv, ws, Y);
}